// GENConvLayer_74371653698181
// MI455X (gfx1250) — compile-verified
//
#include <hip/hip_runtime.h>

typedef __attribute__((ext_vector_type(2))) float v2f;
typedef __attribute__((ext_vector_type(8))) float v8f;

#define N_NODES 10000
#define N_EDGES 320000
#define DIM 256
#define EDGE_DIM 64
#define HEADS 4
#define HEAD_DIM 64
#define FF_HIDDEN 512
#define GAMMA 0.5f

// ---------------------------------------------------------------------------
// f32 WMMA GEMM: D = A(MxK) @ B(KxN).
// Block (256 thr / 8 waves) computes one 16x256 output panel:
//   - A tile (16 x 64 f32, 4 KB) staged in LDS once per K-chunk, shared by all
//     8 waves (ds_load_b64 feeds the WMMA A fragments).
//   - each wave computes a 16x32 strip: two v8f accumulators sharing one A
//     fragment -> 2 v_wmma per ds_load.
// Requires: M % 16 == 0, K % 64 == 0, N % 256 == 0 (true for all calls).
// mode 0: D = acc
// mode 1: D = relu(acc)                       (FF1)
// mode 2: D = resid + relu(addin + acc)       (FF2 + residual wrapper)
// ---------------------------------------------------------------------------
__global__ __launch_bounds__(256) void gemm_f32_wmma(
    const float* __restrict__ A, const float* __restrict__ B,
    float* __restrict__ D, int M, int N, int K, int mode,
    const float* __restrict__ addin, const float* __restrict__ resid)
{
    __shared__ float As[16 * 64];

    const int npanels = N >> 8;                      // 256-wide panels per row
    const int bm = blockIdx.x / npanels;
    const int bn = blockIdx.x - bm * npanels;
    const int m0 = bm << 4;

    const int wv    = threadIdx.x >> 5;              // wave in block: 0..7
    const int lane  = threadIdx.x & 31;
    const int l16   = lane & 15;
    const int khalf = (lane >> 4) << 1;              // 0 or 2

    const int nbase = (bn << 8) + (wv << 5);         // wave's 32-col strip
    const int ncol0 = nbase + l16;
    const int ncol1 = ncol0 + 16;

    // cooperative A-tile load mapping: one float4 per thread
    const int ldrow = threadIdx.x >> 4;              // 0..15
    const int ldcol = (threadIdx.x & 15) << 2;       // 0..60

    v8f acc0 = {}, acc1 = {};

    for (int kc = 0; kc < K; kc += 64) {
        __syncthreads();
        *reinterpret_cast<float4*>(&As[ldrow * 64 + ldcol]) =
            *reinterpret_cast<const float4*>(&A[(long)(m0 + ldrow) * K + kc + ldcol]);
        __syncthreads();

#pragma unroll
        for (int kk = 0; kk < 64; kk += 4) {
            const v2f a = *reinterpret_cast<const v2f*>(&As[l16 * 64 + kk + khalf]);
            const float* Brow = B + (long)(kc + kk + khalf) * N;
            v2f b0, b1;
            b0.x = Brow[ncol0];
            b0.y = Brow[N + ncol0];
            b1.x = Brow[ncol1];
            b1.y = Brow[N + ncol1];
            acc0 = __builtin_amdgcn_wmma_f32_16x16x4_f32(
                false, a, false, b0, (short)0, acc0, false, false);
            acc1 = __builtin_amdgcn_wmma_f32_16x16x4_f32(
                false, a, false, b1, (short)0, acc1, false, false);
        }
    }

    const int rbase = m0 + ((lane >> 4) << 3);       // rows v+0..7 / v+8..15
#pragma unroll
    for (int v = 0; v < 8; ++v) {
        const long i0 = (long)(rbase + v) * N + ncol0;
        const long i1 = (long)(rbase + v) * N + ncol1;
        float v0 = acc0[v], v1 = acc1[v];
        if (mode == 1) {
            v0 = fmaxf(v0, 0.f);
            v1 = fmaxf(v1, 0.f);
        } else if (mode == 2) {
            v0 = resid[i0] + fmaxf(addin[i0] + v0, 0.f);
            v1 = resid[i1] + fmaxf(addin[i1] + v1, 0.f);
        }
        D[i0] = v0;
        D[i1] = v1;
    }
}

// ---------------------------------------------------------------------------
// In-degree: deg[dst] += 1  (atomics land in L2; deg is 40 KB)
// ---------------------------------------------------------------------------
__global__ __launch_bounds__(256) void deg_kernel(
    const int* __restrict__ dst, float* __restrict__ deg, int E)
{
    int e = blockIdx.x * blockDim.x + threadIdx.x;
    if (e < E) atomicAdd(&deg[dst[e]], 1.0f);
}

// ---------------------------------------------------------------------------
// One hop: for each edge, m = relu(h_prev[src] + e_feat), agg[dst] += m.
// Thread = (edge, 4-channel chunk); consecutive threads cover one edge row
// -> fully coalesced float4 reads; atomics hit the 10 MB agg in L2.
// ---------------------------------------------------------------------------
__global__ __launch_bounds__(256) void scatter_kernel(
    const int* __restrict__ src, const int* __restrict__ dst,
    const float* __restrict__ hprev, const float* __restrict__ efeat,
    float* __restrict__ agg, int E)
{
    long gid = (long)blockIdx.x * blockDim.x + threadIdx.x;
    if (gid >= (long)E * (DIM / 4)) return;
    const int e = (int)(gid >> 6);                   // DIM/4 = 64 chunks/edge
    const int c = ((int)gid & 63) << 2;
    const int s = src[e], d = dst[e];

    const float4 h  = *reinterpret_cast<const float4*>(hprev + (long)s * DIM + c);
    const float4 ef = *reinterpret_cast<const float4*>(efeat + (long)e * DIM + c);
    float4 m;
    m.x = fmaxf(h.x + ef.x, 0.f);
    m.y = fmaxf(h.y + ef.y, 0.f);
    m.z = fmaxf(h.z + ef.z, 0.f);
    m.w = fmaxf(h.w + ef.w, 0.f);

    float* ap = agg + (long)d * DIM + c;
    atomicAdd(ap + 0, m.x);
    atomicAdd(ap + 1, m.y);
    atomicAdd(ap + 2, m.z);
    atomicAdd(ap + 3, m.w);
}

// ---------------------------------------------------------------------------
// Graph elimination: h_k -= gamma * deg[row] * h_{k-2}   (in place)
// ---------------------------------------------------------------------------
__global__ __launch_bounds__(256) void elim_kernel(
    float* __restrict__ hk, const float* __restrict__ deg,
    const float* __restrict__ prev2, int total)
{
    int i = blockIdx.x * blockDim.x + threadIdx.x;
    if (i >= total) return;
    hk[i] -= GAMMA * deg[i >> 8] * prev2[i];         // DIM = 256 -> row = i>>8
}

// ---------------------------------------------------------------------------
// Hop attention: one wave per (node, head). Lane owns dims {lane, lane+32}.
// scores_k = sum_d a[h][d] * tanh(hs_k + q);  softmax over 4 hops; blend.
// Butterfly shfl_xor reduction leaves the full sum in every lane (wave32).
// ---------------------------------------------------------------------------
__global__ __launch_bounds__(256) void attn_kernel(
    const float* __restrict__ h0, const float* __restrict__ h1,
    const float* __restrict__ h2, const float* __restrict__ h3,
    const float* __restrict__ att_a, float* __restrict__ out, int N)
{
    const int w    = blockIdx.x * (blockDim.x >> 5) + (threadIdx.x >> 5);
    const int lane = threadIdx.x & 31;
    const int node = w >> 2, head = w & 3;
    if (node >= N) return;

    const float* hs[4] = { h0, h1, h2, h3 };
    const long base = (long)node * DIM + head * HEAD_DIM;
    const float q0 = h0[base + lane], q1 = h0[base + lane + 32];
    const float a0 = att_a[head * HEAD_DIM + lane];
    const float a1 = att_a[head * HEAD_DIM + lane + 32];

    float sc[4], v0[4], v1[4];
#pragma unroll
    for (int k = 0; k < 4; ++k) {
        v0[k] = hs[k][base + lane];
        v1[k] = hs[k][base + lane + 32];
        float s = a0 * tanhf(v0[k] + q0) + a1 * tanhf(v1[k] + q1);
#pragma unroll
        for (int off = 16; off > 0; off >>= 1) s += __shfl_xor(s, off);
        sc[k] = s;
    }
    const float mx = fmaxf(fmaxf(sc[0], sc[1]), fmaxf(sc[2], sc[3]));
    float ex[4], sum = 0.f;
#pragma unroll
    for (int k = 0; k < 4; ++k) { ex[k] = __expf(sc[k] - mx); sum += ex[k]; }
    const float inv = 1.f / sum;
    float o0 = 0.f, o1 = 0.f;
#pragma unroll
    for (int k = 0; k < 4; ++k) {
        o0 += ex[k] * inv * v0[k];
        o1 += ex[k] * inv * v1[k];
    }
    out[base + lane]      = o0;
    out[base + lane + 32] = o1;
}

// ---------------------------------------------------------------------------
extern "C" void kernel_launch(void* const* d_in, const int* in_sizes, int n_in,
                              void* d_out, int out_size, void* d_ws, size_t ws_size,
                              hipStream_t stream)
{
    const float* x         = (const float*)d_in[0];
    const int*   eidx      = (const int*)  d_in[1];   // (2,E): [0:E]=src, [E:2E]=dst
    const float* edge_attr = (const float*)d_in[2];
    const float* W_x       = (const float*)d_in[3];
    const float* W_e       = (const float*)d_in[4];
    const float* att_a     = (const float*)d_in[5];
    const float* W_ff1     = (const float*)d_in[6];
    const float* W_ff2     = (const float*)d_in[7];
    float* out = (float*)d_out;

    const int N = N_NODES, E = N_EDGES;
    const int* src = eidx;
    const int* dst = eidx + E;

    // workspace carve-out (256B aligned)
    char* ws = (char*)d_ws;
    size_t off = 0;
    auto carve = [&](size_t bytes) {
        char* p = ws + off;
        off += (bytes + 255) & ~(size_t)255;
        return p;
    };
    float* h[4];
    for (int k = 0; k < 4; ++k) h[k] = (float*)carve((size_t)N * DIM * 4);
    float* attn   = (float*)carve((size_t)N * DIM * 4);
    float* hidden = (float*)carve((size_t)N * FF_HIDDEN * 4);
    float* deg    = (float*)carve((size_t)N * 4);
    float* efeat  = (float*)carve((size_t)E * DIM * 4);   // 327.7 MB

    // degree
    hipMemsetAsync(deg, 0, (size_t)N * 4, stream);
    deg_kernel<<<(E + 255) / 256, 256, 0, stream>>>(dst, deg, E);

    // h0 = x @ W_x   (10000x256 @ 256x256): one block per 16x256 panel
    gemm_f32_wmma<<<(N / 16) * (DIM / 256), 256, 0, stream>>>(
        x, W_x, h[0], N, DIM, DIM, 0, nullptr, nullptr);

    // e = edge_attr @ W_e  (320000x64 @ 64x256)
    gemm_f32_wmma<<<(E / 16) * (DIM / 256), 256, 0, stream>>>(
        edge_attr, W_e, efeat, E, DIM, EDGE_DIM, 0, nullptr, nullptr);

    // K-hop propagation with elimination
    const int scatter_blocks = (int)(((long)E * (DIM / 4) + 255) / 256);
    for (int k = 1; k <= 3; ++k) {
        hipMemsetAsync(h[k], 0, (size_t)N * DIM * 4, stream);
        scatter_kernel<<<scatter_blocks, 256, 0, stream>>>(
            src, dst, h[k - 1], efeat, h[k], E);
        if (k >= 2)
            elim_kernel<<<(N * DIM + 255) / 256, 256, 0, stream>>>(
                h[k], deg, h[k - 2], N * DIM);
        // k == 1: h_prev2 is all-zero -> no elimination term
    }

    // hop attention -> attn (N x 256);  N*HEADS waves, 8 per block
    attn_kernel<<<(N * HEADS) / 8, 256, 0, stream>>>(
        h[0], h[1], h[2], h[3], att_a, attn, N);

    // FF1: hidden = relu(attn @ W_ff1)   (10000x256 @ 256x512)
    gemm_f32_wmma<<<(N / 16) * (FF_HIDDEN / 256), 256, 0, stream>>>(
        attn, W_ff1, hidden, N, FF_HIDDEN, DIM, 1, nullptr, nullptr);

    // FF2 + wrapper: out = x + relu(attn + hidden @ W_ff2)
    gemm_f32_wmma<<<(N / 16) * (DIM / 256), 256, 0, stream>>>(
        hidden, W_ff2, out, N, DIM, FF_HIDDEN, 2, attn, x);
}